// ChamferDistanceLoss_72043781423454
// MI455X (gfx1250) — compile-verified
//
#include <hip/hip_runtime.h>

typedef float v2f __attribute__((ext_vector_type(2)));
typedef float v8f __attribute__((ext_vector_type(8)));

#define WAVES_PER_BLOCK 8
#define BLOCK_THREADS (WAVES_PER_BLOCK * 32)

// One wave owns TWO 16-point tiles of Q ("query"); loops over all tiles of T
// ("target"); writes sum of per-row min squared distances (unscaled) to
// part[pairIdx].  dist^2 = |q|^2 + |t|^2 - 2 q.t via one exact-fp32 WMMA:
//   A = [-2x_q, -2y_q, -2z_q, 1]   (16x4, -2 folded on constant query side)
//   B = [ x_t,   y_t,   z_t, |t|^2] (4x16)
//   C = |q|^2 per row  ->  D = squared-distance tile.
template <bool GUARD>
__global__ __launch_bounds__(BLOCK_THREADS)
void chamfer_dir_kernel(const float* __restrict__ Q,   // [B, n, 3]
                        const float* __restrict__ T,   // [B, m, 3]
                        float* __restrict__ part,      // [B * ceil(tilesN/2)]
                        int nb, int n, int m)
{
    const int lane = threadIdx.x & 31;
    const int wave = threadIdx.x >> 5;
    const int tilesN = (n + 15) >> 4;
    const int tilesM = (m + 15) >> 4;
    const int pairs  = (tilesN + 1) >> 1;     // 2 query tiles per wave

    const int tp = blockIdx.x * WAVES_PER_BLOCK + wave;
    const int b  = tp / pairs;                // wave-uniform
    const int p  = tp % pairs;
    if (b >= nb) return;                      // uniform exit: EXEC stays full

    const bool hi  = lane >= 16;
    const int  sub = lane & 15;

    // ---- load the two A tiles (32 query points) ----
    const float* qb_batch = Q + (size_t)b * n * 3;
    int row0 = p * 32 + sub;
    int row1 = row0 + 16;
    int r0c = row0 < n ? row0 : (n - 1);
    int r1c = row1 < n ? row1 : (n - 1);

    const float* q0 = qb_batch + (size_t)r0c * 3;
    const float* q1 = qb_batch + (size_t)r1c * 3;
    float x0 = q0[0], y0 = q0[1], z0 = q0[2];
    float x1 = q1[0], y1 = q1[1], z1 = q1[2];
    float pn0 = x0 * x0 + y0 * y0 + z0 * z0;
    float pn1 = x1 * x1 + y1 * y1 + z1 * z1;

    v2f A0, A1;
    A0.x = hi ? (-2.0f * z0) : (-2.0f * x0);
    A0.y = hi ? 1.0f         : (-2.0f * y0);
    A1.x = hi ? (-2.0f * z1) : (-2.0f * x1);
    A1.y = hi ? 1.0f         : (-2.0f * y1);

    // ---- C matrices: C[r] = |q|^2 of row (r + 8*hi) ----
    v8f C0, C1;
#pragma unroll
    for (int r = 0; r < 8; ++r) {
        C0[r] = __shfl(pn0, r + (hi ? 8 : 0), 32);
        C1[r] = __shfl(pn1, r + (hi ? 8 : 0), 32);
    }

    v8f m0, m1;
#pragma unroll
    for (int r = 0; r < 8; ++r) { m0[r] = 3.0e38f; m1[r] = 3.0e38f; }

    const float* tb = T + ((size_t)b * m + sub) * 3;   // lane's column base

    for (int j = 0; j < tilesM; ++j) {
        float tx, ty, tz, tn;
        if (GUARD) {
            int  gc = j * 16 + sub;
            bool ok = gc < m;
            const float* gb = T + ((size_t)b * m + (ok ? gc : 0)) * 3;
            tx = gb[0]; ty = gb[1]; tz = gb[2];
            float tnn = tx * tx + ty * ty + tz * tz;
            tx = ok ? tx : 0.0f;
            ty = ok ? ty : 0.0f;
            tz = ok ? tz : 0.0f;
            tn = ok ? tnn : 3.0e37f;
        } else {
            const float* gb = tb + (size_t)j * 48;     // 16 points * 3 floats
            tx = gb[0]; ty = gb[1]; tz = gb[2];
            tn = tx * tx + ty * ty + tz * tz;
        }

        v2f Bv;
        Bv.x = hi ? tz : tx;
        Bv.y = hi ? tn : ty;

        v8f D0 = __builtin_amdgcn_wmma_f32_16x16x4_f32(
            false, A0, false, Bv, (short)0, C0, false, false);
        v8f D1 = __builtin_amdgcn_wmma_f32_16x16x4_f32(
            false, A1, false, Bv, (short)0, C1, false, false);

#pragma unroll
        for (int r = 0; r < 8; ++r) {
            m0[r] = fminf(m0[r], D0[r]);
            m1[r] = fminf(m1[r], D1[r]);
        }
    }

    // ---- min over the 16 columns within each half-wave ----
#pragma unroll
    for (int off = 1; off < 16; off <<= 1) {
#pragma unroll
        for (int r = 0; r < 8; ++r) {
            m0[r] = fminf(m0[r], __shfl_xor(m0[r], off, 32));
            m1[r] = fminf(m1[r], __shfl_xor(m1[r], off, 32));
        }
    }

    // ---- sum valid rows (rows r+0..7 in lo half, r+8..15 in hi half) ----
    float s = 0.0f;
#pragma unroll
    for (int r = 0; r < 8; ++r) {
        int rr0 = p * 32 + r + (hi ? 8 : 0);
        int rr1 = rr0 + 16;
        if (rr0 < n) s += m0[r];
        if (rr1 < n) s += m1[r];
    }
    float total = __shfl(s, 0, 32) + __shfl(s, 16, 32);
    if (lane == 0) part[tp] = total;
}

// Deterministic fixed-order final reduction of both directions' partials.
__global__ __launch_bounds__(256)
void chamfer_reduce_kernel(const float* __restrict__ ws, float* __restrict__ out,
                           int n1, int n2, float s1, float s2)
{
    __shared__ float sm[256];
    float a = 0.0f, b = 0.0f;
    for (int i = threadIdx.x; i < n1; i += 256) a += ws[i];
    for (int i = threadIdx.x; i < n2; i += 256) b += ws[n1 + i];
    sm[threadIdx.x] = a * s1 + b * s2;
    __syncthreads();
    for (int off = 128; off > 0; off >>= 1) {
        if ((int)threadIdx.x < off) sm[threadIdx.x] += sm[threadIdx.x + off];
        __syncthreads();
    }
    if (threadIdx.x == 0) out[0] = sm[0];
}

extern "C" void kernel_launch(void* const* d_in, const int* in_sizes, int n_in,
                              void* d_out, int out_size, void* d_ws, size_t ws_size,
                              hipStream_t stream)
{
    (void)n_in; (void)out_size; (void)ws_size;
    const float* pred = (const float*)d_in[0];
    const float* gt   = (const float*)d_in[1];
    float* out = (float*)d_out;
    float* ws  = (float*)d_ws;

    const int B = 4;
    const int N = in_sizes[0] / (B * 3);
    const int M = in_sizes[1] / (B * 3);

    const int tilesN = (N + 15) / 16;
    const int tilesM = (M + 15) / 16;
    const int pairsN = (tilesN + 1) / 2;
    const int pairsM = (tilesM + 1) / 2;
    const int waves0 = B * pairsN;                  // pred -> gt
    const int waves1 = B * pairsM;                  // gt -> pred
    const int blocks0 = (waves0 + WAVES_PER_BLOCK - 1) / WAVES_PER_BLOCK;
    const int blocks1 = (waves1 + WAVES_PER_BLOCK - 1) / WAVES_PER_BLOCK;

    float* part0 = ws;                              // [waves0]
    float* part1 = ws + waves0;                     // [waves1]

    const bool aligned = (N % 16 == 0) && (M % 16 == 0);
    if (aligned) {
        chamfer_dir_kernel<false><<<blocks0, BLOCK_THREADS, 0, stream>>>(
            pred, gt, part0, B, N, M);
        chamfer_dir_kernel<false><<<blocks1, BLOCK_THREADS, 0, stream>>>(
            gt, pred, part1, B, M, N);
    } else {
        chamfer_dir_kernel<true><<<blocks0, BLOCK_THREADS, 0, stream>>>(
            pred, gt, part0, B, N, M);
        chamfer_dir_kernel<true><<<blocks1, BLOCK_THREADS, 0, stream>>>(
            gt, pred, part1, B, M, N);
    }

    chamfer_reduce_kernel<<<1, 256, 0, stream>>>(
        ws, out, waves0, waves1,
        1.0f / (float)(B * N), 1.0f / (float)(B * M));
}